// ExamplarLoss_6262062317976
// MI455X (gfx1250) — compile-verified
//
#include <hip/hip_runtime.h>
#include <math.h>

// Problem constants from the reference
#define BATCH   4096
#define DDIM    128
#define NROWS   (2 * BATCH)   // 8192
#define EPS_F   1.0f

typedef float v2f __attribute__((ext_vector_type(2)));
typedef float v8f __attribute__((ext_vector_type(8)));

// ---------------------------------------------------------------------------
// Kernel 1: interleave + L2-normalize rows into workspace.
// Row 2i   = latent1[i] / (sqrt(eps) * ||latent1[i]||)
// Row 2i+1 = latent2[i] / (sqrt(eps) * ||latent2[i]||)
// Then the Gram matrix of the normalized rows IS the cosine-sim logits.
// Also zero the global loss accumulator (must happen every launch).
// ---------------------------------------------------------------------------
__global__ void __launch_bounds__(DDIM)
normalize_kernel(const float* __restrict__ l1, const float* __restrict__ l2,
                 float* __restrict__ lnorm, float* __restrict__ acc) {
  const int row = blockIdx.x;           // 0..NROWS-1
  const int tid = threadIdx.x;          // 0..127
  const float* src = (row & 1) ? l2 : l1;
  const float v = src[(size_t)(row >> 1) * DDIM + tid];

  // sum of squares: wave32 shuffle reduce, then cross-wave via LDS
  float sq = v * v;
  #pragma unroll
  for (int m = 16; m >= 1; m >>= 1) sq += __shfl_xor(sq, m, 32);

  __shared__ float partial[4];
  __shared__ float scale_s;
  const int wave = tid >> 5;
  if ((tid & 31) == 0) partial[wave] = sq;
  __syncthreads();
  if (tid == 0) {
    const float tot = partial[0] + partial[1] + partial[2] + partial[3];
    scale_s = 1.0f / (sqrtf(EPS_F) * sqrtf(tot));
    if (row == 0) acc[0] = 0.0f;        // re-init accumulator every launch
  }
  __syncthreads();
  lnorm[(size_t)row * DDIM + tid] = v * scale_s;
}

// ---------------------------------------------------------------------------
// Kernel 2: fused Gram + flash log-softmax. One wave32 per 16-row strip.
//
// V_WMMA_F32_16X16X4_F32 operand layout (wave32):
//   A 16x4 : lane holds A[m = lane%16, k] with k = 2*(lane/16) + {vgpr0,vgpr1}
//   B 4x16 : symmetric (n = lane%16, same k mapping)
// -> for S = Ln * Ln^T both fragments are a float2 load at
//    Ln[row, 4*chunk + 2*(lane/16)].
//   C/D 16x16: vgpr v, lane l -> (M = v + 8*(l/16), N = l%16)
// ---------------------------------------------------------------------------
__global__ void __launch_bounds__(32)
exemplar_main_kernel(const float* __restrict__ lnorm, float* __restrict__ acc) {
  const int lane = threadIdx.x & 31;
  const int half = lane >> 4;           // 0 or 1 (K sub-block)
  const int n16  = lane & 15;
  const int row0 = blockIdx.x * 16;     // strip of 16 rows

  // Preload A strip fragments: 32 chunks of K=4 -> 64 VGPRs
  v2f a[DDIM / 4];
  const float* arow = lnorm + (size_t)(row0 + n16) * DDIM + 2 * half;
  #pragma unroll
  for (int c = 0; c < DDIM / 4; ++c)
    a[c] = *(const v2f*)(arow + 4 * c);

  // Per-lane online-softmax state for the 8 (row,col) cells this lane owns
  float mx[8], sm[8], ps[8];
  #pragma unroll
  for (int v = 0; v < 8; ++v) { mx[v] = -INFINITY; sm[v] = 0.0f; ps[v] = 0.0f; }

  for (int t = 0; t < NROWS / 16; ++t) {
    const int col0 = t * 16;
    const float* brow = lnorm + (size_t)(col0 + n16) * DDIM + 2 * half;

    v8f c = {};
    #pragma unroll
    for (int k = 0; k < DDIM / 4; ++k) {
      const v2f b = *(const v2f*)(brow + 4 * k);
      // D = A x B + C, fp32 matrix core
      c = __builtin_amdgcn_wmma_f32_16x16x4_f32(
          /*neg_a=*/false, a[k], /*neg_b=*/false, b,
          /*c_mod=*/(short)0, c, /*reuse_a=*/false, /*reuse_b=*/false);
    }

    const int gc = col0 + n16;
    #pragma unroll
    for (int v = 0; v < 8; ++v) {
      const int gr = row0 + v + 8 * half;
      float s = c[v];
      if (gc == gr) s = 0.0f;                 // (1 - eye) mask: diagonal -> 0
      if (gc == (gr ^ 1)) ps[v] = s;          // positive logit for this row
      const float om = mx[v];
      const float nm = fmaxf(om, s);
      sm[v] = sm[v] * __expf(om - nm) + __expf(s - nm);
      mx[v] = nm;
    }
  }

  // Merge (max, sumexp, pos) across the 16 lanes that share each row.
  // xor masks 1,2,4,8 stay within lane groups [0,15] and [16,31].
  #pragma unroll
  for (int v = 0; v < 8; ++v) {
    float m = mx[v], s = sm[v], p = ps[v];
    #pragma unroll
    for (int d = 8; d >= 1; d >>= 1) {
      const float m2 = __shfl_xor(m, d, 32);
      const float s2 = __shfl_xor(s, d, 32);
      const float nm = fmaxf(m, m2);
      s = s * __expf(m - nm) + s2 * __expf(m2 - nm);
      m = nm;
      p += __shfl_xor(p, d, 32);              // exactly one lane is nonzero
    }
    mx[v] = m; sm[v] = s; ps[v] = p;
  }

  // Lanes 0 and 16 each own 8 complete rows: logp_pos = pos - max - log(sumexp)
  if (n16 == 0) {
    float local = 0.0f;
    #pragma unroll
    for (int v = 0; v < 8; ++v)
      local += ps[v] - mx[v] - __logf(sm[v]);
    atomicAdd(acc, local);                    // 1024 atomics total: negligible
  }
}

// ---------------------------------------------------------------------------
// Kernel 3: loss = -sum(logp_pos) / N
// ---------------------------------------------------------------------------
__global__ void finalize_kernel(const float* __restrict__ acc,
                                float* __restrict__ out) {
  out[0] = -acc[0] / (float)NROWS;
}

// ---------------------------------------------------------------------------
extern "C" void kernel_launch(void* const* d_in, const int* in_sizes, int n_in,
                              void* d_out, int out_size, void* d_ws, size_t ws_size,
                              hipStream_t stream) {
  const float* latent1 = (const float*)d_in[0];
  const float* latent2 = (const float*)d_in[1];
  float* out = (float*)d_out;

  // Workspace layout: [0..3] loss accumulator, [256..] normalized matrix (4 MB)
  float* acc   = (float*)d_ws;
  float* lnorm = (float*)((char*)d_ws + 256);

  normalize_kernel<<<NROWS, DDIM, 0, stream>>>(latent1, latent2, lnorm, acc);
  exemplar_main_kernel<<<NROWS / 16, 32, 0, stream>>>(lnorm, acc);
  finalize_kernel<<<1, 1, 0, stream>>>(acc, out);
}